// FilterDelayNetwork_57380763074715
// MI455X (gfx1250) — compile-verified
//
#include <hip/hip_runtime.h>
#include <hip/hip_bf16.h>

typedef __attribute__((ext_vector_type(2))) float v2f;
typedef __attribute__((ext_vector_type(4))) float v4f;
typedef __attribute__((ext_vector_type(8))) float v8f;

#if __has_builtin(__builtin_amdgcn_global_load_async_to_lds_b32)
#define FDN_ASYNC 1
#else
#define FDN_ASYNC 0
#endif

#define FDN_DELAY 1499
#define WIN 272    // 256 outputs + 16-sample halo
#define PADW 288   // 17 rows at stride 17 (max index 17*16+15 = 287)

typedef __attribute__((address_space(1))) int g_int;
typedef __attribute__((address_space(3))) int l_int;

// z[n] = sum_{k=0}^{16} h2[k] * x[n - 1499 - k]
// Per-wave tile of 256 outputs: Z[i][j] = z[tile*256 + 16*j + i]
// Z = T_cur*C + T_prev*P   (8 x v_wmma_f32_16x16x4_f32)
// Window x[base-16 .. base+255] staged into padded LDS (stride-17 rows, bank-conflict-free),
// via async global->LDS copies, double-buffered and software-pipelined across tiles.

__global__ __launch_bounds__(256) void fdn_fir_wmma_kernel(const float* __restrict__ X,
                                                           float* __restrict__ Z, int n)
{
    __shared__ float h2s[20];
    __shared__ float sbuf[8][2][PADW];

    if (threadIdx.x == 0) {
        // Exact reference constant derivation (float64, cast to f32 as reference does)
        double g   = pow(10.0, -3.0 * 1499.0 / 48000.0 / 2.0);
        double p   = log10(10.0 / 4.0) * (1.0 - 1.0 / (0.7 * 0.7)) * log10(g);
        float  pf  = (float)p;
        float  b0f = (float)(g * (1.0 - p));
        double beta   = (1.0 - 0.7) / (1.0 + 0.7);
        double inv1mb = 1.0 / (1.0 - beta);
        h2s[0] = (float)((double)b0f * inv1mb);
        double pw = 1.0;
        for (int k = 1; k <= 15; ++k) {
            h2s[k] = (float)((double)b0f * pw * ((double)pf - beta) * inv1mb);
            pw *= (double)pf;
        }
        h2s[16] = (float)(-beta * (double)b0f * pw * inv1mb);
        h2s[17] = 0.0f; h2s[18] = 0.0f; h2s[19] = 0.0f;
    }
    __syncthreads();

    const int lane = threadIdx.x & 31;
    const int wv   = threadIdx.x >> 5;
    const int lid  = lane & 15;          // A row / B,D column
    const int th2  = (lane >> 4) << 1;   // K sub-slot pair: 0 (lanes 0-15) or 2 (lanes 16-31)

    auto tap = [&](int lag) -> float {
        return (lag >= 0 && lag <= 16) ? h2s[lag] : 0.0f;
    };

    // Filter matrices in VGPRs (A operand, 32-bit A 16x4 layout)
    v2f Ac[4], Ap[4];
#pragma unroll
    for (int q = 0; q < 4; ++q) {
        const int t0 = 4 * q + th2;
        Ac[q][0] = tap(lid - t0);
        Ac[q][1] = tap(lid - t0 - 1);
        Ap[q][0] = tap(lid - t0 + 16);
        Ap[q][1] = tap(lid - t0 + 15);
    }

    const int tiles = n >> 8;
    const int wavesPerBlock = blockDim.x >> 5;
    const int gw = blockIdx.x * wavesPerBlock + ((int)threadIdx.x >> 5);
    const int nw = gridDim.x * wavesPerBlock;

    // ---- staging: window x[base-16 .. base+255] -> padded LDS (sidx = 17*(w>>4) + (w&15))
    auto stage_async = [&](int tile, int b) {
#if FDN_ASYNC
        const int base16 = (tile << 8) - FDN_DELAY - 16;
        float* S = &sbuf[wv][b][0];
#pragma unroll
        for (int k = 0; k < 9; ++k) {
            const int w = 32 * k + lane;
            if (w < WIN) {
                const int sidx = ((w >> 4) * 17) + (w & 15);
                __builtin_amdgcn_global_load_async_to_lds_b32(
                    (g_int*)(X + base16 + w),
                    (l_int*)(S + sidx), 0, 0);
            }
        }
#else
        (void)tile; (void)b;
#endif
    };

    auto stage_guard = [&](int tile, int b) {   // boundary tiles: zero-pad below index 0
        const int base16 = (tile << 8) - FDN_DELAY - 16;
        float* S = &sbuf[wv][b][0];
#pragma unroll
        for (int k = 0; k < 9; ++k) {
            const int w = 32 * k + lane;
            if (w < WIN) {
                const int g = base16 + w;
                const int sidx = ((w >> 4) * 17) + (w & 15);
                S[sidx] = (g >= 0) ? X[g] : 0.0f;
            }
        }
    };

    // ---- consume: 8 WMMAs from LDS, store 256 outputs (two b128 per lane)
    auto consume = [&](int tile, int b) {
        const float* S = &sbuf[wv][b][0];
        const int scur = 17 * (lid + 1) + th2;   // current 16-block row
        const int sprv = 17 * lid + th2;         // previous 16-block (halo) row
        v8f acc = {0.0f, 0.0f, 0.0f, 0.0f, 0.0f, 0.0f, 0.0f, 0.0f};
#pragma unroll
        for (int q = 0; q < 4; ++q) {
            v2f bc, bp;
            bc[0] = S[scur + 4 * q];
            bc[1] = S[scur + 4 * q + 1];
            bp[0] = S[sprv + 4 * q];
            bp[1] = S[sprv + 4 * q + 1];
            acc = __builtin_amdgcn_wmma_f32_16x16x4_f32(
                false, Ac[q], false, bc, (short)0, acc, false, false);
            acc = __builtin_amdgcn_wmma_f32_16x16x4_f32(
                false, Ap[q], false, bp, (short)0, acc, false, false);
        }
        const int o = (tile << 8) + (lid << 4) + (th2 << 2);
        v4f lo = {acc[0], acc[1], acc[2], acc[3]};
        v4f hi = {acc[4], acc[5], acc[6], acc[7]};
        *(v4f*)(Z + o)     = lo;
        *(v4f*)(Z + o + 4) = hi;
    };

#if FDN_ASYNC
    int tile = gw;
    if (tile < tiles) {
        int cb = 0;
        if (tile >= 6) stage_async(tile, cb); else stage_guard(tile, cb);
        for (; tile < tiles; tile += nw) {
            const int  nxt  = tile + nw;
            const bool hasn = (nxt < tiles);
            if (hasn) {
                if (nxt >= 6) stage_async(nxt, cb ^ 1); else stage_guard(nxt, cb ^ 1);
            }
            // async loads complete in order: <=9 outstanding drains current tile's 9
            if (hasn && nxt >= 6) {
                asm volatile("s_wait_asynccnt 0x9" ::: "memory");
            } else {
                asm volatile("s_wait_asynccnt 0x0" ::: "memory");
            }
            consume(tile, cb);
            cb ^= 1;
        }
    }
#else
    for (int tile = gw; tile < tiles; tile += nw) {
        stage_guard(tile, 0);
        asm volatile("s_wait_dscnt 0x0" ::: "memory");   // LDS per-wave is in-order; belt & braces
        consume(tile, 0);
    }
#endif
}

extern "C" void kernel_launch(void* const* d_in, const int* in_sizes, int n_in,
                              void* d_out, int out_size, void* d_ws, size_t ws_size,
                              hipStream_t stream) {
    (void)n_in; (void)out_size; (void)d_ws; (void)ws_size;
    const float* X = (const float*)d_in[0];
    float*       Z = (float*)d_out;
    const int n = in_sizes[0];   // 2^24, multiple of 256
    fdn_fir_wmma_kernel<<<dim3(1024), dim3(256), 0, stream>>>(X, Z, n);
}